// GCN_RED_84335977825023
// MI455X (gfx1250) — compile-verified
//
#include <hip/hip_runtime.h>

typedef float v2f __attribute__((ext_vector_type(2)));
typedef float v8f __attribute__((ext_vector_type(8)));

#define FD 64  // feature dim

// One wave = one 16-edge x 64-feature tile.
// r = review_feat @ W^T via v_wmma_f32_16x16x4_f32, then fused
// gather/scale/scatter-atomic into out (pre-zeroed).
__global__ __launch_bounds__(256) void gcn_edge_wmma_kernel(
    const float* __restrict__ feature,      // [N, 64]
    const float* __restrict__ review_feat,  // [E, 64]
    const float* __restrict__ ci,           // [N]
    const float* __restrict__ W,            // [64, 64] row-major (torch Linear weight)
    const float* __restrict__ drop_mask,    // [E]
    const int*   __restrict__ src_idx,      // [E]
    const int*   __restrict__ dst_idx,      // [E]
    float*       __restrict__ out,          // [N, 64] accumulator (zeroed)
    int n_edges, int n_tiles)
{
    // W^T staged in B-fragment layout: frag f = ks*4 + nb holds
    // B[k = ks*4 .. ks*4+3][n = nb*16 .. nb*16+15], B[k][n] = W[n][k].
    // Fragment lane mapping (32-bit B 4x16): lane = (k>>1)*16 + n%16,
    // vgpr v = k&1  ->  lane stores float2 {W[n][k0], W[n][k0+1]}, k0 = base + (lane>>4)*2.
    __shared__ v2f lw[64 * 32];

    const int tid = threadIdx.x;
    for (int idx = tid; idx < 64 * 32; idx += 256) {
        int frag = idx >> 5;
        int lane = idx & 31;
        int ks = frag >> 2;
        int nb = frag & 3;
        int n  = nb * 16 + (lane & 15);
        int k  = ks * 4 + ((lane >> 4) << 1);
        v2f w;
        w.x = W[n * FD + k];
        w.y = W[n * FD + k + 1];
        lw[idx] = w;
    }
    __syncthreads();

    const int lane = tid & 31;
    const int wave = tid >> 5;
    const int tile = blockIdx.x * 8 + wave;
    if (tile >= n_tiles) return;   // wave-uniform; EXEC stays all-ones for WMMA
    const int e0 = tile * 16;

    // A fragment (16x4 f32): lane holds row M = lane%16, K pair base (lane>>4)*2.
    int arow = e0 + (lane & 15);
    if (arow >= n_edges) arow = n_edges - 1;           // clamp: epilogue skips these
    const float* aptr = review_feat + (size_t)arow * FD + ((lane >> 4) << 1);

    v8f acc0 = {}, acc1 = {}, acc2 = {}, acc3 = {};

    #pragma unroll
    for (int ks = 0; ks < 16; ++ks) {
        v2f a  = *(const v2f*)(aptr + ks * 4);          // global_load_b64
        v2f b0 = lw[(ks * 4 + 0) * 32 + lane];          // ds_load_b64, conflict-free
        v2f b1 = lw[(ks * 4 + 1) * 32 + lane];
        v2f b2 = lw[(ks * 4 + 2) * 32 + lane];
        v2f b3 = lw[(ks * 4 + 3) * 32 + lane];
        acc0 = __builtin_amdgcn_wmma_f32_16x16x4_f32(false, a, false, b0, (short)0, acc0, false, false);
        acc1 = __builtin_amdgcn_wmma_f32_16x16x4_f32(false, a, false, b1, (short)0, acc1, false, false);
        acc2 = __builtin_amdgcn_wmma_f32_16x16x4_f32(false, a, false, b2, (short)0, acc2, false, false);
        acc3 = __builtin_amdgcn_wmma_f32_16x16x4_f32(false, a, false, b3, (short)0, acc3, false, false);
    }

    // C/D layout: vgpr p, lane-group g -> edge m = g*8 + p, column n = lane%16 (+ nb*16).
    const int lg = lane >> 4;
    const int ln = lane & 15;
    #pragma unroll
    for (int p = 0; p < 8; ++p) {
        int e = e0 + lg * 8 + p;
        if (e >= n_edges) continue;
        int s = src_idx[e];
        int d = dst_idx[e];
        float cs = ci[s] * drop_mask[e];
        const float* frow = feature + (size_t)s * FD;   // coalesced per 16-lane group
        float* orow = out + (size_t)d * FD;
        atomicAdd(&orow[ 0 + ln], (acc0[p] + frow[ 0 + ln]) * cs);
        atomicAdd(&orow[16 + ln], (acc1[p] + frow[16 + ln]) * cs);
        atomicAdd(&orow[32 + ln], (acc2[p] + frow[32 + ln]) * cs);
        atomicAdd(&orow[48 + ln], (acc3[p] + frow[48 + ln]) * cs);
    }
}

// out[n, f] *= ci[n]
__global__ __launch_bounds__(256) void gcn_scale_kernel(
    float* __restrict__ out, const float* __restrict__ ci, int total)
{
    int i = blockIdx.x * 256 + threadIdx.x;
    if (i < total) out[i] *= ci[i >> 6];
}

extern "C" void kernel_launch(void* const* d_in, const int* in_sizes, int n_in,
                              void* d_out, int out_size, void* d_ws, size_t ws_size,
                              hipStream_t stream) {
    const float* feature     = (const float*)d_in[0];
    const float* review_feat = (const float*)d_in[1];
    const float* ci          = (const float*)d_in[2];
    const float* W           = (const float*)d_in[3];
    const float* drop_mask   = (const float*)d_in[4];
    const int*   src_idx     = (const int*)d_in[5];
    const int*   dst_idx     = (const int*)d_in[6];
    float* out = (float*)d_out;

    const int E = in_sizes[1] / FD;

    // Zero the accumulator (graph-capturable memset node).
    hipMemsetAsync(d_out, 0, (size_t)out_size * sizeof(float), stream);

    const int n_tiles = (E + 15) / 16;
    const int blocks  = (n_tiles + 7) / 8;   // 8 waves (tiles) per 256-thread block
    gcn_edge_wmma_kernel<<<blocks, 256, 0, stream>>>(
        feature, review_feat, ci, W, drop_mask, src_idx, dst_idx, out, E, n_tiles);

    gcn_scale_kernel<<<(out_size + 255) / 256, 256, 0, stream>>>(out, ci, out_size);
}